// PPRIteration_10213432229943
// MI455X (gfx1250) — compile-verified
//
#include <hip/hip_runtime.h>
#include <hip/hip_bf16.h>
#include <stdint.h>

#define RESTART_P   0.1f
#define D_FEAT      64
#define EPC         8                       // edges (rows) gathered per TDM op (max for 32-bit idx)
#define WPB         8                       // wave32s per block (256 threads)
#define BUF_FLOATS  (EPC * D_FEAT)          // 512 floats = 2 KB per LDS buffer
#define CHUNKS_PER_WAVE 4

typedef __attribute__((ext_vector_type(4))) unsigned int v4u_;
typedef __attribute__((ext_vector_type(8))) int          v8i_;
typedef __attribute__((ext_vector_type(4))) int          v4i_;

// ---------------------------------------------------------------------------
// TDM gather: pull up to 8 feature rows (row length 64 f32) into LDS.
// D# per CDNA5 ISA (gather mode, 32-bit indices).
// 6-arg builtin form (clang-23 / therock-10.0 lane).
// ---------------------------------------------------------------------------
__device__ __forceinline__ void tdm_gather_rows(
    const float* feat, uint32_t lds_byte_addr, int n_nodes, int nvalid,
    int r0, int r1, int r2, int r3, int r4, int r5, int r6, int r7) {
  uint64_t ga = (uint64_t)(uintptr_t)feat;

  v4u_ g0;
  g0.x = 1u | (1u << 30) | (1u << 31);          // count=1, idx_size=32b, gather_mode=1
  g0.y = lds_byte_addr;                         // lds_addr
  g0.z = (uint32_t)ga;                          // global_addr[31:0]
  g0.w = (uint32_t)((ga >> 32) & 0x1FFFFFFu)    // global_addr[56:32]
       | (2u << 30);                            // type = 2 ("image")

  v8i_ g1;
  g1[0] = (int)(2u << 16);                                      // data_size=4B; wg_mask=0
  g1[1] = (int)((uint32_t)D_FEAT << 16);                        // tensor_dim0 lo16 = 64
  g1[2] = (int)(((uint32_t)n_nodes & 0xFFFFu) << 16);           // dim0 hi=0 | tensor_dim1 lo16
  g1[3] = (int)((((uint32_t)n_nodes >> 16) & 0xFFFFu)           // tensor_dim1 hi16
       | ((uint32_t)D_FEAT << 16));                             // tile_dim0 = 64
  g1[4] = (int)((uint32_t)nvalid & 0xFFFFu);                    // tile_dim1 = #valid indices
  g1[5] = D_FEAT;                                               // tensor_dim0_stride lo32 = 64
  g1[6] = 0;                                                    // dim0_stride hi | dim1_stride lo
  g1[7] = 0;

  v4i_ g2; g2.x = r0; g2.y = r1; g2.z = r2; g2.w = r3;          // row_index_0..3
  v4i_ g3; g3.x = r4; g3.y = r5; g3.z = r6; g3.w = r7;          // row_index_4..7

  v8i_ gx;                                                      // extra operand (6-arg form)
  gx[0] = 0; gx[1] = 0; gx[2] = 0; gx[3] = 0;
  gx[4] = 0; gx[5] = 0; gx[6] = 0; gx[7] = 0;

  __builtin_amdgcn_tensor_load_to_lds(g0, g1, g2, g3, gx, 0);
}

// ---------------------------------------------------------------------------
// Kernel (emitted FIRST so the disasm snippet shows the TDM + atomic path):
// COO scatter-add.  out[dst,:] += 0.9 * w_e * features[src,:]
// One wave32 per chunk of 8 edges, TDM-gathered rows, double-buffered LDS.
// ---------------------------------------------------------------------------
__global__ __launch_bounds__(WPB * 32) void ppr_scatter_kernel(
    const float* __restrict__ feat, const float* __restrict__ ew,
    const int*  __restrict__ esrc, const int*  __restrict__ edst,
    float* __restrict__ out, int n_edges, int n_nodes) {

  __shared__ float smem[WPB * 2 * BUF_FLOATS];

  const int lane   = (int)(threadIdx.x & 31u);
  const int wib    = (int)(threadIdx.x >> 5);
  const int gwave  = (int)blockIdx.x * WPB + wib;
  const int nwaves = (int)gridDim.x * WPB;
  const int n_chunks = (n_edges + EPC - 1) / EPC;

  float* wbuf0 = &smem[wib * 2 * BUF_FLOATS];
  // Flat LDS aperture address: low 32 bits are the LDS byte offset (ISA §10.2).
  const uint32_t lds0 = (uint32_t)(uintptr_t)(void*)wbuf0;
  const float scale = 1.0f - RESTART_P;

  int chunk = gwave;
  if (chunk >= n_chunks) return;   // wave-uniform: idle waves never touch the TDM

  auto load_edges = [&](int c, int& s, float& w, int& d, int& nv) {
    int e0 = c * EPC;
    nv = n_edges - e0; if (nv > EPC) nv = EPC;
    int e = e0 + lane;
    s = 0; w = 0.0f; d = 0;
    if (lane < EPC && e < n_edges) { s = esrc[e]; w = ew[e]; d = edst[e]; }
  };

  auto issue = [&](int parity, int s, int nv) {
    tdm_gather_rows(feat, lds0 + (uint32_t)parity * (BUF_FLOATS * 4), n_nodes, nv,
                    __builtin_amdgcn_readlane(s, 0),
                    __builtin_amdgcn_readlane(s, 1),
                    __builtin_amdgcn_readlane(s, 2),
                    __builtin_amdgcn_readlane(s, 3),
                    __builtin_amdgcn_readlane(s, 4),
                    __builtin_amdgcn_readlane(s, 5),
                    __builtin_amdgcn_readlane(s, 6),
                    __builtin_amdgcn_readlane(s, 7));
  };

  // Prologue: fill buffer 0.
  int s, d, nv; float w;
  load_edges(chunk, s, w, d, nv);
  issue(0, s, nv);

  int parity = 0;
  for (;;) {
    int  next     = chunk + nwaves;
    bool has_next = next < n_chunks;
    int ns = 0, nd = 0, nnv = 0; float nw = 0.0f;

    if (has_next) {
      load_edges(next, ns, nw, nd, nnv);
      issue(parity ^ 1, ns, nnv);                 // overlap next DMA with compute
      __builtin_amdgcn_s_wait_tensorcnt(1);       // in-order => current buffer ready
    } else {
      __builtin_amdgcn_s_wait_tensorcnt(0);
    }

    const float* buf = wbuf0 + parity * BUF_FLOATS;
    for (int i = 0; i < nv; ++i) {
      float wi = __shfl(w, i, 32) * scale;        // uniform per edge
      int   di = __shfl(d, i, 32);
      float2 f = *(const float2*)(buf + i * D_FEAT + lane * 2);
      float* o = out + (size_t)(uint32_t)di * D_FEAT + lane * 2;
      unsafeAtomicAdd(o,     f.x * wi);           // global_atomic_add_f32
      unsafeAtomicAdd(o + 1, f.y * wi);
    }

    if (!has_next) break;
    chunk = next; parity ^= 1;
    s = ns; w = nw; d = nd; nv = nnv;
  }
}

// ---------------------------------------------------------------------------
// out = RESTART_P * H0   (accumulator init, float4 vectorized)
// ---------------------------------------------------------------------------
__global__ void ppr_init_kernel(const float4* __restrict__ h0,
                                float4* __restrict__ out, int n4) {
  int i = blockIdx.x * blockDim.x + threadIdx.x;
  if (i < n4) {
    float4 v = h0[i];
    out[i] = make_float4(v.x * RESTART_P, v.y * RESTART_P,
                         v.z * RESTART_P, v.w * RESTART_P);
  }
}

// ---------------------------------------------------------------------------
// out = max(out, 0)  (float4 vectorized, in place)
// ---------------------------------------------------------------------------
__global__ void ppr_relu_kernel(float4* __restrict__ out, int n4) {
  int i = blockIdx.x * blockDim.x + threadIdx.x;
  if (i < n4) {
    float4 v = out[i];
    v.x = fmaxf(v.x, 0.0f);
    v.y = fmaxf(v.y, 0.0f);
    v.z = fmaxf(v.z, 0.0f);
    v.w = fmaxf(v.w, 0.0f);
    out[i] = v;
  }
}

// ---------------------------------------------------------------------------
// Host entry
// ---------------------------------------------------------------------------
extern "C" void kernel_launch(void* const* d_in, const int* in_sizes, int n_in,
                              void* d_out, int out_size, void* d_ws, size_t ws_size,
                              hipStream_t stream) {
  const float* feat = (const float*)d_in[0];
  const float* h0   = (const float*)d_in[1];
  const float* ew   = (const float*)d_in[2];
  const int*   esrc = (const int*)d_in[3];
  const int*   edst = (const int*)d_in[4];
  float* out = (float*)d_out;

  const int n_nodes = in_sizes[0] / D_FEAT;
  const int n_edges = in_sizes[2];
  const int n4 = out_size / 4;

  {
    int blocks = (n4 + 255) / 256;
    ppr_init_kernel<<<blocks, 256, 0, stream>>>((const float4*)h0,
                                                (float4*)out, n4);
  }
  {
    int n_chunks = (n_edges + EPC - 1) / EPC;
    int waves  = (n_chunks + CHUNKS_PER_WAVE - 1) / CHUNKS_PER_WAVE;
    int blocks = (waves + WPB - 1) / WPB;
    if (blocks < 1) blocks = 1;
    ppr_scatter_kernel<<<blocks, WPB * 32, 0, stream>>>(feat, ew, esrc, edst,
                                                        out, n_edges, n_nodes);
  }
  {
    int blocks = (n4 + 255) / 256;
    ppr_relu_kernel<<<blocks, 256, 0, stream>>>((float4*)out, n4);
  }
}